// LGMLoss_v0_76897094468407
// MI455X (gfx1250) — compile-verified
//
#include <hip/hip_runtime.h>

typedef __attribute__((ext_vector_type(2))) float v2f;
typedef __attribute__((ext_vector_type(8))) float v8f;

#define ALPHA      0.1f
#define NCLASS     10000
#define DFEAT      256
#define NBATCH     8192

#define TM   128   // block tile M
#define TN   64    // block tile N
#define TK   32    // K chunk staged in LDS
#define LDSP 36    // padded LDS row stride (floats): 144B rows -> 16B-aligned for
                   // async B128 writes AND conflict-free b64 operand reads

// Async copy of 16 bytes global -> LDS (GLOBAL_LOAD_ASYNC_TO_LDS_B128, ASYNCcnt).
// LDS address = low 32 bits of the generic pointer (ISA 10.2: LDS aperture uses addr[31:0]).
__device__ __forceinline__ void async_copy16(const float* g, float* l) {
  unsigned lds = (unsigned)(unsigned long long)l;
  unsigned long long ga = (unsigned long long)g;
  asm volatile("global_load_async_to_lds_b128 %0, %1, off"
               :: "v"(lds), "v"(ga) : "memory");
}

__device__ __forceinline__ void wait_async0() {
  asm volatile("s_wait_asynccnt 0x0" ::: "memory");
}

// ---------------------------------------------------------------------------
// Kernel 1: squared row norms of feat (rows 0..B-1) and centers (rows B..B+C-1)
// ---------------------------------------------------------------------------
__global__ __launch_bounds__(256) void lgm_sqnorm_kernel(
    const float* __restrict__ feat, const float* __restrict__ centers,
    float* __restrict__ fsq, float* __restrict__ csq) {
  const int wid  = threadIdx.x >> 5;
  const int lane = threadIdx.x & 31;
  const int n = blockIdx.x * 8 + wid;

  const float* src;
  float* dst;
  if (n < NBATCH) {
    src = feat + (size_t)n * DFEAT;               dst = fsq + n;
  } else if (n < NBATCH + NCLASS) {
    src = centers + (size_t)(n - NBATCH) * DFEAT; dst = csq + (n - NBATCH);
  } else {
    return;
  }
  float s = 0.f;
#pragma unroll
  for (int j = 0; j < DFEAT / 32; ++j) {
    float v = src[lane + 32 * j];
    s += v * v;
  }
#pragma unroll
  for (int off = 16; off > 0; off >>= 1) s += __shfl_xor(s, off, 32);
  if (lane == 0) *dst = s;
}

// ---------------------------------------------------------------------------
// Kernel 2: fused GEMM (cross = feat @ centers^T) + LGM epilogue via WMMA f32,
// double-buffered async global->LDS staging, non-temporal output streaming.
// ---------------------------------------------------------------------------
__global__ __launch_bounds__(256) void lgm_gemm_kernel(
    const float* __restrict__ feat, const float* __restrict__ centers,
    const int*   __restrict__ label,
    const float* __restrict__ fsq,  const float* __restrict__ csq,
    float* __restrict__ out_logits, float* __restrict__ out_margin) {

  __shared__ float smem[2 * (TM + TN) * LDSP];   // double-buffered A|B tiles

  const int tid  = threadIdx.x;
  const int wid  = tid >> 5;        // 0..7 -> owns 16 M-rows
  const int lane = tid & 31;
  const int m16  = lane & 15;       // row-in-tile (A) / col-in-tile (B)
  const int khi  = lane >> 4;       // selects K pair {0,1} vs {2,3}

  const int blockM = blockIdx.y * TM;
  const int blockN = blockIdx.x * TN;

  v8f acc[4] = {};                  // four 16x16 f32 accumulators (16 x 64 strip)

  // Issue one K-stage of async copies: A = 128x32 f32 (1024 16B chunks),
  // B = 64x32 f32 (512 16B chunks), spread across 8 waves x 32 lanes.
  auto issue_stage = [&](int k0, int buf) {
    float* Abase = smem + buf * (TM + TN) * LDSP;
    float* Bbase = Abase + TM * LDSP;
#pragma unroll
    for (int i = 0; i < 4; ++i) {
      const int chunk = wid * 128 + i * 32 + lane;   // 0..1023
      const int row = chunk >> 3;                    // 0..127
      const int c4  = (chunk & 7) * 4;               // 0,4,..,28
      async_copy16(feat + (size_t)(blockM + row) * DFEAT + k0 + c4,
                   Abase + row * LDSP + c4);
    }
#pragma unroll
    for (int i = 0; i < 2; ++i) {
      const int chunk = wid * 64 + i * 32 + lane;    // 0..511
      const int row = chunk >> 3;                    // 0..63
      const int c4  = (chunk & 7) * 4;
      int cB = blockN + row;
      if (cB >= NCLASS) cB = NCLASS - 1;             // clamp; stores guarded later
      async_copy16(centers + (size_t)cB * DFEAT + k0 + c4,
                   Bbase + row * LDSP + c4);
    }
  };

  // ---- prologue: fill buffer 0 ----
  issue_stage(0, 0);
  wait_async0();
  __syncthreads();

  int cur = 0;
  for (int k0 = 0; k0 < DFEAT; k0 += TK) {
    // overlap: start next stage into the other buffer
    if (k0 + TK < DFEAT) issue_stage(k0 + TK, cur ^ 1);

    const float* Abase = smem + cur * (TM + TN) * LDSP;
    const float* Bbase = Abase + TM * LDSP;
    const float* pa = Abase + (wid * 16 + m16) * LDSP + khi * 2;

#pragma unroll
    for (int kk = 0; kk < TK; kk += 4) {
      v2f av = *(const v2f*)(pa + kk);
#pragma unroll
      for (int nt = 0; nt < 4; ++nt) {
        const float* pb = Bbase + (nt * 16 + m16) * LDSP + khi * 2 + kk;
        v2f bv = *(const v2f*)pb;
        acc[nt] = __builtin_amdgcn_wmma_f32_16x16x4_f32(
            /*neg_a=*/false, av, /*neg_b=*/false, bv,
            /*c_mod=*/(short)0, acc[nt], /*reuse_a=*/false, /*reuse_b=*/false);
      }
    }

    wait_async0();       // my async copies for the next stage have landed
    __syncthreads();     // everyone's have; current buffer free for reuse
    cur ^= 1;
  }

  // ---- fused epilogue: dist = |f|^2 + |c|^2 - 2*cross; logits & margin ----
  // D layout: vgpr r, lane l -> row = r + 8*(l>>4), col = l&15
  const int rowBase = blockM + wid * 16 + 8 * khi;
  float fsq_r[8];
  int   lab_r[8];
#pragma unroll
  for (int r = 0; r < 8; ++r) {
    const int row = rowBase + r;
    fsq_r[r] = fsq[row];
    lab_r[r] = label[row];
  }

#pragma unroll
  for (int nt = 0; nt < 4; ++nt) {
    const int col = blockN + nt * 16 + m16;
    const bool valid = (col < NCLASS);
    const float cs = valid ? csq[col] : 0.f;
#pragma unroll
    for (int r = 0; r < 8; ++r) {
      const int row = rowBase + r;
      const float dist = fsq_r[r] + cs - 2.0f * acc[nt][r];
      const float lg = -0.5f * dist;
      const float mg = (lab_r[r] == col) ? lg * (1.0f + ALPHA) : lg;
      if (valid) {
        const size_t idx = (size_t)row * NCLASS + col;
        // Outputs are streamed once and never re-read: non-temporal stores keep
        // the (fully L2-resident) feat/centers working set hot in the 192MB L2.
        __builtin_nontemporal_store(lg, out_logits + idx);
        __builtin_nontemporal_store(mg, out_margin + idx);
      }
    }
  }
}

// ---------------------------------------------------------------------------
// Kernel 3a: per-row squared distance to the labeled center
// ---------------------------------------------------------------------------
__global__ __launch_bounds__(256) void lgm_rowlike_kernel(
    const float* __restrict__ feat, const float* __restrict__ centers,
    const int* __restrict__ label, float* __restrict__ rowlike) {
  const int wid  = threadIdx.x >> 5;
  const int lane = threadIdx.x & 31;
  const int row = blockIdx.x * 8 + wid;
  if (row >= NBATCH) return;
  int lab = label[row];
  if (lab < 0) lab = 0;
  if (lab >= NCLASS) lab = NCLASS - 1;
  const float* f = feat + (size_t)row * DFEAT;
  const float* c = centers + (size_t)lab * DFEAT;
  float s = 0.f;
#pragma unroll
  for (int j = 0; j < DFEAT / 32; ++j) {
    const float d = f[lane + 32 * j] - c[lane + 32 * j];
    s += d * d;
  }
#pragma unroll
  for (int off = 16; off > 0; off >>= 1) s += __shfl_xor(s, off, 32);
  if (lane == 0) rowlike[row] = s;
}

// ---------------------------------------------------------------------------
// Kernel 3b: deterministic single-block reduction -> likelihood scalar
// ---------------------------------------------------------------------------
__global__ __launch_bounds__(256) void lgm_reduce_kernel(
    const float* __restrict__ rowlike, float* __restrict__ likeSlot) {
  __shared__ float red[8];
  const int tid  = threadIdx.x;
  const int wid  = tid >> 5;
  const int lane = tid & 31;
  float s = 0.f;
#pragma unroll
  for (int j = 0; j < NBATCH / 256; ++j) s += rowlike[tid + 256 * j];
#pragma unroll
  for (int off = 16; off > 0; off >>= 1) s += __shfl_xor(s, off, 32);
  if (lane == 0) red[wid] = s;
  __syncthreads();
  if (tid == 0) {
    float t = 0.f;
#pragma unroll
    for (int w = 0; w < 8; ++w) t += red[w];
    *likeSlot = t * (1.0f / (2.0f * (float)NBATCH));
  }
}

// ---------------------------------------------------------------------------
extern "C" void kernel_launch(void* const* d_in, const int* in_sizes, int n_in,
                              void* d_out, int out_size, void* d_ws, size_t ws_size,
                              hipStream_t stream) {
  const float* feat    = (const float*)d_in[0];
  const int*   label   = (const int*)d_in[1];
  const float* centers = (const float*)d_in[2];

  float* out        = (float*)d_out;
  float* out_logits = out;
  float* out_margin = out + (size_t)NBATCH * NCLASS;
  float* likeSlot   = out + 2 * (size_t)NBATCH * NCLASS;

  float* fsq     = (float*)d_ws;            // [NBATCH]
  float* csq     = fsq + NBATCH;            // [NCLASS]
  float* rowlike = csq + NCLASS;            // [NBATCH]

  // 1) row norms
  {
    const int rows = NBATCH + NCLASS;
    lgm_sqnorm_kernel<<<(rows + 7) / 8, 256, 0, stream>>>(feat, centers, fsq, csq);
  }
  // 2) fused GEMM + epilogue
  {
    dim3 grid((NCLASS + TN - 1) / TN, NBATCH / TM);
    lgm_gemm_kernel<<<grid, 256, 0, stream>>>(feat, centers, label, fsq, csq,
                                              out_logits, out_margin);
  }
  // 3) likelihood (deterministic two-stage)
  lgm_rowlike_kernel<<<NBATCH / 8, 256, 0, stream>>>(feat, centers, label, rowlike);
  lgm_reduce_kernel<<<1, 256, 0, stream>>>(rowlike, likeSlot);
}